// DictionaryMatchingTv_32822140076389
// MI455X (gfx1250) — compile-verified
//
#include <hip/hip_runtime.h>
#include <math.h>

// Problem constants (from reference)
#define ETL    16
#define DATOMS 4000
#define NPIX   (192 * 192)

typedef __attribute__((ext_vector_type(2))) float v2f;
typedef __attribute__((ext_vector_type(4))) float v4f;
typedef __attribute__((ext_vector_type(8))) float v8f;

// ---------------------------------------------------------------------------
// Prep 1: db = safe_normalize(db_mag * se_mask)
//         d2h[d] = -0.5 * sum(db*db)   (WMMA C-matrix bias; argmin -> argmax)
// ---------------------------------------------------------------------------
__global__ __launch_bounds__(256) void prep_db_kernel(
    const float* __restrict__ db_mag, const float* __restrict__ dt_ms,
    float* __restrict__ dbn, float* __restrict__ d2h) {
  int d = blockIdx.x * blockDim.x + threadIdx.x;
  if (d >= DATOMS) return;
  float v[ETL];
  float nsq = 0.f;
#pragma unroll
  for (int e = 0; e < ETL; ++e) {
    float m = (dt_ms[e] * 1e-3f < 1e-3f) ? 1.f : 0.f;  // SE-echo mask
    float x = db_mag[(size_t)d * ETL + e] * m;
    v[e] = x;
    nsq = fmaf(x, x, nsq);
  }
  float nrm = sqrtf(nsq);
  float ss = 0.f;
  if (nrm > 0.f) {
#pragma unroll
    for (int e = 0; e < ETL; ++e) {
      float y = v[e] / nrm;                 // matches jnp norm-then-divide
      dbn[(size_t)d * ETL + e] = y;
      ss = fmaf(y, y, ss);
    }
  } else {
#pragma unroll
    for (int e = 0; e < ETL; ++e) dbn[(size_t)d * ETL + e] = 0.f;  // nan_to_num
  }
  d2h[d] = -0.5f * ss;
}

// ---------------------------------------------------------------------------
// Prep 2: s = safe_normalize(safe_normalize(sig) * se_mask); s2[n] = sum(s*s)
// ---------------------------------------------------------------------------
__global__ __launch_bounds__(256) void prep_sig_kernel(
    const float* __restrict__ sig, const float* __restrict__ dt_ms,
    float* __restrict__ sn, float* __restrict__ s2) {
  int n = blockIdx.x * blockDim.x + threadIdx.x;
  if (n >= NPIX) return;
  float v[ETL];
  float nsq = 0.f;
#pragma unroll
  for (int e = 0; e < ETL; ++e) {
    float x = sig[(size_t)n * ETL + e];
    v[e] = x;
    nsq = fmaf(x, x, nsq);
  }
  float nrm1 = sqrtf(nsq);                   // first normalize: all 16 echoes
  float msq = 0.f;
#pragma unroll
  for (int e = 0; e < ETL; ++e) {
    float t = (nrm1 > 0.f) ? v[e] / nrm1 : 0.f;
    float m = (dt_ms[e] * 1e-3f < 1e-3f) ? 1.f : 0.f;
    float x = t * m;                         // mask, then normalize again
    v[e] = x;
    msq = fmaf(x, x, msq);
  }
  float nrm2 = sqrtf(msq);
  float ss = 0.f;
  if (nrm2 > 0.f) {
#pragma unroll
    for (int e = 0; e < ETL; ++e) {
      float y = v[e] / nrm2;
      sn[(size_t)n * ETL + e] = y;
      ss = fmaf(y, y, ss);
    }
  } else {
#pragma unroll
    for (int e = 0; e < ETL; ++e) sn[(size_t)n * ETL + e] = 0.f;
  }
  s2[n] = ss;
}

// ---------------------------------------------------------------------------
// Main fused GEMM + argmin. One wave32 owns 16 pixels, sweeps all 4000 atoms
// in 16-row tiles using chained V_WMMA_F32_16X16X4_F32 (K=16 -> 4 WMMAs/tile).
// The -0.5*||db||^2 bias is loaded straight into the C accumulator, so after
// the WMMA chain c[r] = dot - 0.5*d2 and argmin(dist) == argmax(c): the hot
// loop is pure compare/select (no per-tile fma correction needed).
//
// f32 WMMA layouts (ISA 7.12.2):
//   A 16x4  : lanes 0-15 -> M=lane, VGPR{0,1}=K{0,1}; lanes 16-31 -> K{2,3}
//   B 4x16  : lanes 0-15 -> N=lane, VGPR{0,1}=K{0,1}; lanes 16-31 -> K{2,3}
//   C 16x16 : lane l, vgpr r -> (M = 8*(l>=16)+r, N = l&15)
// ---------------------------------------------------------------------------
__global__ __launch_bounds__(256) void match_kernel(
    const float* __restrict__ dbn, const float* __restrict__ d2h,
    const float* __restrict__ sn, const float* __restrict__ s2,
    const float* __restrict__ t2s, const float* __restrict__ b1s,
    float* __restrict__ out_t2, float* __restrict__ out_b1,
    float* __restrict__ out_dist) {
  int gtid = blockIdx.x * blockDim.x + threadIdx.x;
  int wave = gtid >> 5;        // wave32
  int lane = gtid & 31;
  int p0 = wave * 16;          // 16 pixels per wave
  if (p0 >= NPIX) return;      // wave-uniform; EXEC stays all-ones for WMMA
  int hf = lane >> 4;          // which K/M half this lane serves
  int lo = lane & 15;

  // Loop-invariant B fragments: this wave's 16 normalized signal vectors.
  v2f bf[4];
#pragma unroll
  for (int ch = 0; ch < 4; ++ch)
    bf[ch] = *(const v2f*)(sn + (size_t)(p0 + lo) * ETL + 4 * ch + 2 * hf);

  float bestC = -3.0e38f;  // maximize c = dot - 0.5*d2  (== minimize dist)
  int bestI = 0;

  for (int d0 = 0; d0 < DATOMS; d0 += 16) {
    // C init = -0.5*d2 for this lane's 8 atom rows (two b128 broadcasts).
    v4f ca = *(const v4f*)(d2h + d0 + 8 * hf);
    v4f cb = *(const v4f*)(d2h + d0 + 8 * hf + 4);
    v8f c = {ca[0], ca[1], ca[2], ca[3], cb[0], cb[1], cb[2], cb[3]};
#pragma unroll
    for (int ch = 0; ch < 4; ++ch) {
      v2f a = *(const v2f*)(dbn + (size_t)(d0 + lo) * ETL + 4 * ch + 2 * hf);
      c = __builtin_amdgcn_wmma_f32_16x16x4_f32(
          /*neg_a=*/false, a, /*neg_b=*/false, bf[ch],
          /*c_mod=*/(short)0, c, /*reuse_a=*/false, /*reuse_b=*/false);
    }
    int base = d0 + 8 * hf;
#pragma unroll
    for (int r = 0; r < 8; ++r) {
      // strict > keeps the earliest index within this lane's atom set,
      // matching jnp.argmin first-occurrence semantics.
      if (c[r] > bestC) { bestC = c[r]; bestI = base + r; }
    }
  }

  // Merge the two M-halves: lane l vs lane l^16 hold the same pixel column.
  float oc = __shfl_xor(bestC, 16, 32);
  int oi = __shfl_xor(bestI, 16, 32);
  if (oc > bestC || (oc == bestC && oi < bestI)) { bestC = oc; bestI = oi; }

  if (hf == 0) {
    int p = p0 + lo;
    out_t2[p] = t2s[bestI];
    out_b1[p] = b1s[bestI];
    // dist = sqrt(max(d2 + s2 - 2*dot, 0)) = sqrt(max(s2 - 2*c, 0))
    out_dist[p] = sqrtf(fmaxf(fmaf(-2.0f, bestC, s2[p]), 0.0f));
  }
}

// ---------------------------------------------------------------------------
// Launch wrapper
// Inputs: 0 slice_signal[36864*16], 1 db_mag[4000*16], 2 db_t2s_s[4000],
//         3 db_b1s[4000], 4 delta_t_r2p_ms[16]
// Output: [t2(36864) | b1(36864) | min_dist(36864)] f32
// ---------------------------------------------------------------------------
extern "C" void kernel_launch(void* const* d_in, const int* in_sizes, int n_in,
                              void* d_out, int out_size, void* d_ws,
                              size_t ws_size, hipStream_t stream) {
  const float* sig = (const float*)d_in[0];
  const float* dbm = (const float*)d_in[1];
  const float* t2s = (const float*)d_in[2];
  const float* b1s = (const float*)d_in[3];
  const float* dt = (const float*)d_in[4];

  float* ws = (float*)d_ws;
  float* dbn = ws;                         // 64000 f32  (normalized dictionary)
  float* d2h = ws + 64000;                 // 4000  f32  (-0.5*||db||^2)
  float* sn = ws + 68000;                  // 589824 f32 (normalized signal)
  float* s2 = ws + 657824;                 // 36864 f32
  // total ws use: ~2.78 MB

  float* out = (float*)d_out;
  float* out_t2 = out;
  float* out_b1 = out + NPIX;
  float* out_dist = out + 2 * NPIX;

  prep_db_kernel<<<(DATOMS + 255) / 256, 256, 0, stream>>>(dbm, dt, dbn, d2h);
  prep_sig_kernel<<<(NPIX + 255) / 256, 256, 0, stream>>>(sig, dt, sn, s2);

  // 36864/16 = 2304 waves -> 73728 threads -> 288 blocks of 256 (8 waves/WG)
  int nwaves = NPIX / 16;
  int nthreads = nwaves * 32;
  match_kernel<<<nthreads / 256, 256, 0, stream>>>(dbn, d2h, sn, s2, t2s, b1s,
                                                   out_t2, out_b1, out_dist);
}